// SparseDenseMatMul_37443524887287
// MI455X (gfx1250) — compile-verified
//
#include <hip/hip_runtime.h>

typedef __attribute__((ext_vector_type(2))) float v2f;
typedef __attribute__((ext_vector_type(4))) float v4f;
typedef __attribute__((ext_vector_type(8))) float v8f;

// ---------------------------------------------------------------------------
// Zero the output (harness poisons d_out with 0xAA before timing).
// ---------------------------------------------------------------------------
__global__ void zero_f32x4_kernel(v4f* __restrict__ out, long n4) {
    long i = (long)blockIdx.x * blockDim.x + threadIdx.x;
    long stride = (long)gridDim.x * blockDim.x;
    v4f z = {0.0f, 0.0f, 0.0f, 0.0f};
    for (; i < n4; i += stride) out[i] = z;
}

// ---------------------------------------------------------------------------
// COO SpMM, one wave per batch of 16 nonzeros.
//   For a batch: G[k, :] = A[cols[k], :]   (16 x 64 gathered rows)
//                C       = diag(vals) * G  (16 x 64, via WMMA f32 16x16x4)
//   then C row m is atomically scattered into out[rows[m], :].
//
// WMMA mapping (wave32, fp32):
//   A-operand 16x4: lane L -> M = L&15, K = 2*(L>=16) + vgpr   (diag slice)
//   B-operand 4x16: lane L -> N = L&15, K = 2*(L>=16) + vgpr   (gathered A)
//   C/D 16x16:      vgpr g, lane L -> M = g + 8*(L>=16), N = L&15
// K=16 is realized by chaining 4 WMMAs (kk = 0..3), accumulating in C.
//
// All 32 B-gathers for a batch are issued up-front (tile-major) so in-order
// load returns let the scheduler overlap gathers with the WMMA chains and
// atomic scatters via partial s_wait_loadcnt.
// All addressing uses 32-bit ELEMENT offsets against the uniform base
// pointers so VMEM lowers to the SGPR-base + 32-bit-voffset `scale_offset`
// form (no per-lane 64-bit address arithmetic); the tile displacement
// (tile*64 bytes) folds into the instruction immediate offset.
// ---------------------------------------------------------------------------
__global__ void spmm_coo_wmma_kernel(const float* __restrict__ vals,
                                     const float* __restrict__ A,
                                     const int*   __restrict__ rows,
                                     const int*   __restrict__ cols,
                                     float*       __restrict__ out,
                                     int nbatch) {
    const int lane  = threadIdx.x & 31;
    const int m     = lane & 15;            // M (and N) sub-index for this lane
    const int khalf = (lane >> 4) << 1;     // 0 for lanes 0-15, 2 for lanes 16-31

    const int wave   = (int)((blockIdx.x * blockDim.x + threadIdx.x) >> 5);
    const int nwaves = (int)((gridDim.x * blockDim.x) >> 5);

    for (int b = wave; b < nbatch; b += nwaves) {
        const int li = (b << 4) + m;        // lanes 16-31 mirror lanes 0-15
        const float v  = vals[li];
        const int   r  = rows[li];
        const int   cb = cols[li] << 6;     // element offset of row cols[li] in A

        // 32-bit gather offsets (already displaced by this lane's column m).
        int bx[4], by[4];
#pragma unroll
        for (int kk = 0; kk < 4; ++kk) {
            const int k0 = (kk << 2) + khalf;
            bx[kk] = __shfl(cb, k0,     32) + m;
            by[kk] = __shfl(cb, k0 + 1, 32) + m;
        }

        // 32-bit scatter offsets for the 8 C rows this lane owns
        // (M = g + 8*(lane>=16)).
        int oo[8];
#pragma unroll
        for (int g = 0; g < 8; ++g) {
            const int rM = __shfl(r, g + ((lane >> 4) << 3), 32);
            oo[g] = (rM << 6) + m;
        }

        // Issue ALL 32 B-gathers up front, tile-major so earlier tiles'
        // data returns first (vmem loads complete in order).
        v2f B[4][4];
#pragma unroll
        for (int tile = 0; tile < 4; ++tile) {
#pragma unroll
            for (int kk = 0; kk < 4; ++kk) {
                B[tile][kk].x = A[bx[kk] + (tile << 4)];
                B[tile][kk].y = A[by[kk] + (tile << 4)];
            }
        }

        // Diagonal A-operand slices: A[m, k] = (k == m) ? v_m : 0.
        v2f Aop[4];
#pragma unroll
        for (int kk = 0; kk < 4; ++kk) {
            const int k0 = (kk << 2) + khalf;
            Aop[kk].x = (k0     == m) ? v : 0.0f;
            Aop[kk].y = (k0 + 1 == m) ? v : 0.0f;
        }

        // 4 column tiles of 16; chain K=16 as 4 WMMAs accumulating in C,
        // then scatter that tile while later tiles' WMMAs/loads overlap.
#pragma unroll
        for (int tile = 0; tile < 4; ++tile) {
            v8f acc = {0.0f, 0.0f, 0.0f, 0.0f, 0.0f, 0.0f, 0.0f, 0.0f};
#pragma unroll
            for (int kk = 0; kk < 4; ++kk) {
                acc = __builtin_amdgcn_wmma_f32_16x16x4_f32(
                        /*neg_a=*/false, Aop[kk],
                        /*neg_b=*/false, B[tile][kk],
                        /*c_mod=*/(short)0, acc,
                        /*reuse_a=*/false, /*reuse_b=*/false);
            }
#pragma unroll
            for (int g = 0; g < 8; ++g) {
                // hardware fp32 scatter-atomic into L2-resident out
                unsafeAtomicAdd(&out[oo[g] + (tile << 4)], acc[g]);
            }
        }
    }
}

// ---------------------------------------------------------------------------
// Scalar tail (nnz % 16 != 0): one wave per nonzero, 2 columns per lane.
// ---------------------------------------------------------------------------
__global__ void spmm_coo_tail_kernel(const float* __restrict__ vals,
                                     const float* __restrict__ A,
                                     const int*   __restrict__ rows,
                                     const int*   __restrict__ cols,
                                     float*       __restrict__ out,
                                     int start, int nnz) {
    const int wave = (int)((blockIdx.x * blockDim.x + threadIdx.x) >> 5);
    const int lane = threadIdx.x & 31;
    const int i = start + wave;
    if (i >= nnz) return;
    const float v = vals[i];
    const int   r = rows[i];
    const int   c = cols[i];
    const int   n0 = lane << 1;
    const float a0 = A[(c << 6) + n0];
    const float a1 = A[(c << 6) + n0 + 1];
    unsafeAtomicAdd(&out[(r << 6) + n0],     v * a0);
    unsafeAtomicAdd(&out[(r << 6) + n0 + 1], v * a1);
}

// ---------------------------------------------------------------------------
// Launch
// ---------------------------------------------------------------------------
extern "C" void kernel_launch(void* const* d_in, const int* in_sizes, int n_in,
                              void* d_out, int out_size, void* d_ws, size_t ws_size,
                              hipStream_t stream) {
    const float* vals = (const float*)d_in[0];
    const float* A    = (const float*)d_in[1];
    const int*   rows = (const int*)d_in[2];
    const int*   cols = (const int*)d_in[3];
    float*       out  = (float*)d_out;
    const int    nnz  = in_sizes[0];

    // 1) zero output
    {
        const long n4 = (long)out_size / 4;   // N=64 -> divisible by 4
        int blocks = (int)((n4 + 255) / 256);
        if (blocks > 16384) blocks = 16384;
        zero_f32x4_kernel<<<blocks, 256, 0, stream>>>((v4f*)out, n4);
    }

    // 2) main WMMA COO kernel: one wave per 16-nnz batch, 8 waves per block
    const int nbatch = nnz >> 4;
    if (nbatch > 0) {
        int blocks = (nbatch + 7) / 8;
        spmm_coo_wmma_kernel<<<blocks, 256, 0, stream>>>(vals, A, rows, cols, out, nbatch);
    }

    // 3) tail (not taken for NNZ = 1.6M, but kept for generality)
    const int rem = nnz & 15;
    if (rem > 0) {
        const int start = nbatch << 4;
        const int blocks = (rem * 32 + 255) / 256;
        spmm_coo_tail_kernel<<<blocks, 256, 0, stream>>>(vals, A, rows, cols, out, start, nnz);
    }
}